// CombineRadialNeighSpeciesAndCentralSpecies_48060684042918
// MI455X (gfx1250) — compile-verified
//
#include <hip/hip_runtime.h>

typedef float v2f __attribute__((ext_vector_type(2)));
typedef float v8f __attribute__((ext_vector_type(8)));

#define N_COMP  64
#define D_IN    128
#define D_OUT   128
#define P_PITCH 288   // dwords per K-pair row: 256 data + 32 pad (==32 mod 64 banks)

// One block per sample: out[n] = x[n] (64x128) @ W[species[n]] (128x128), fp32.
// 8 wave32s: wave -> (mt in 0..3, ng in 0..1); each wave owns a 16x64 strip
// (4 accumulator tiles), K swept 4 at a time with V_WMMA_F32_16X16X4_F32.
//
// W[s] is staged into LDS once per block with gfx1250 async global->LDS
// copies (ASYNCcnt), stored K-pair interleaved: element (k,q) at dword
// (k>>1)*P_PITCH + 2q + (k&1). A WMMA B fragment is then one naturally
// paired, bank-conflict-free ds_load_b64 (no register repacking).
__global__ __launch_bounds__(256) void
grouped_species_gemm_kernel(const float* __restrict__ x,
                            const int*   __restrict__ species,
                            const float* __restrict__ W,
                            float*       __restrict__ out)
{
    __shared__ float lds_w[(D_IN / 2) * P_PITCH];   // 72 KB

    const int n    = blockIdx.x;
    const int lane = threadIdx.x;        // 0..31
    const int wave = threadIdx.y;        // 0..7
    const int mt   = wave & 3;           // M tile (rows mt*16..+15)
    const int ng   = wave >> 2;          // N group (cols ng*64..+63)
    const int half = lane >> 4;          // lane 0-15 vs 16-31
    const int l16  = lane & 15;

    const int s = species[n];
    const float* __restrict__ Wb = W + (size_t)s * (D_IN * D_OUT);

    // ---- Stage W[s] (64 KB) into LDS, K-pair interleaved -------------------
    // Thread tid owns column q = tid&127, parity h = tid>>7, rows d = 2i+h.
    // Global side: 32 consecutive floats per wave per op (coalesced 128B).
    // LDS side: stride-2 dwords -> 32 distinct even banks, conflict-free.
    {
        const int tid = wave * 32 + lane;
        const int q   = tid & 127;
        const int h   = tid >> 7;
        const float* gsrc = Wb + (size_t)h * D_OUT + q;          // d = h at i=0
        const unsigned lds_base = (unsigned)(size_t)(&lds_w[0]); // low 32b = LDS off
        const unsigned laddr0 = lds_base + (unsigned)(2 * q + h) * 4u;
#pragma unroll
        for (int i = 0; i < 64; ++i) {
            const unsigned laddr = laddr0 + (unsigned)i * (P_PITCH * 4u);
            const float*   gaddr = gsrc + (size_t)i * (2 * D_OUT);
            asm volatile("global_load_async_to_lds_b32 %0, %1, off"
                         :: "v"(laddr), "v"(gaddr) : "memory");
        }
        asm volatile("s_wait_asynccnt 0x0" ::: "memory");
    }
    __syncthreads();

    // ---- Main GEMM loop ----------------------------------------------------
    // A fragment: row (mt*16 + l16) of x[n]; lane-half picks the K pair.
    const float* __restrict__ xb =
        x + (size_t)n * (N_COMP * D_IN) + (size_t)(mt * 16 + l16) * D_IN;
    const int col0 = ng * 64 + l16;      // this lane's B column for tile 0

    v8f acc0 = {}, acc1 = {}, acc2 = {}, acc3 = {};

#pragma unroll 4
    for (int k0 = 0; k0 < D_IN; k0 += 4) {
        const int kb = k0 + 2 * half;    // this lane-half's K base (even)

        // A: 16x4 f32 tile per ISA 7.12.2 -> one aligned b64 load per lane.
        v2f a = *(const v2f*)(xb + kb);

        // B: K-pair (kb, kb+1) is contiguous in LDS -> one ds_load_b64 per
        // tile, even-aligned dest pair, all 64 dwords on distinct banks.
        const int pb = (kb >> 1) * P_PITCH + 2 * col0;
        v2f b0 = *(const v2f*)(&lds_w[pb +  0]);
        v2f b1 = *(const v2f*)(&lds_w[pb + 32]);
        v2f b2 = *(const v2f*)(&lds_w[pb + 64]);
        v2f b3 = *(const v2f*)(&lds_w[pb + 96]);

        acc0 = __builtin_amdgcn_wmma_f32_16x16x4_f32(false, a, false, b0,
                                                     (short)0, acc0, false, false);
        acc1 = __builtin_amdgcn_wmma_f32_16x16x4_f32(false, a, false, b1,
                                                     (short)0, acc1, false, false);
        acc2 = __builtin_amdgcn_wmma_f32_16x16x4_f32(false, a, false, b2,
                                                     (short)0, acc2, false, false);
        acc3 = __builtin_amdgcn_wmma_f32_16x16x4_f32(false, a, false, b3,
                                                     (short)0, acc3, false, false);
    }

    // C/D layout per ISA 7.12.2: VGPR i holds row (i + 8*half), col l16.
    float* __restrict__ ob = out + (size_t)n * (N_COMP * D_OUT)
                           + (size_t)(mt * 16 + 8 * half) * D_OUT
                           + (ng * 64 + l16);
#pragma unroll
    for (int i = 0; i < 8; ++i) {
        ob[(size_t)i * D_OUT +  0] = acc0[i];
        ob[(size_t)i * D_OUT + 16] = acc1[i];
        ob[(size_t)i * D_OUT + 32] = acc2[i];
        ob[(size_t)i * D_OUT + 48] = acc3[i];
    }
}

extern "C" void kernel_launch(void* const* d_in, const int* in_sizes, int n_in,
                              void* d_out, int out_size, void* d_ws, size_t ws_size,
                              hipStream_t stream)
{
    const float* x       = (const float*)d_in[0];
    const int*   species = (const int*)  d_in[1];
    const float* W       = (const float*)d_in[2];
    float*       out     = (float*)d_out;

    const int n_samples = in_sizes[1];   // one species id per sample

    dim3 grid(n_samples);
    dim3 block(32, 8);                   // 8 wave32s
    hipLaunchKernelGGL(grouped_species_gemm_kernel, grid, block, 0, stream,
                       x, species, W, out);
}